// PtModule_76166950028114
// MI455X (gfx1250) — compile-verified
//
#include <hip/hip_runtime.h>
#include <hip/hip_bf16.h>

// Elementwise: out = (0.75*x + 0.25)^2   [ == (((x+2)*3-5)/4)^2 ]
// Memory-bound (AI ~0.25 FLOP/B): floor = 1.07 GB / 23.3 TB/s ~ 46 us.
// MI455X strategy:
//  - B128 vector loads/stores (widest VMEM path, 16 B/lane/instr).
//  - Non-temporal hints on both sides: 1.07 GB stream, zero reuse, working
//    set >> 192 MB L2 -> don't pollute caches (ISA TH=NT policy).
//  - wave32: 256-thread blocks = 8 waves per block.
// Note: use a native Clang ext_vector float4 — HIP's float4 is a class type
// and __builtin_nontemporal_* only accepts scalar/vector-of-scalar pointers.

typedef float v4f __attribute__((ext_vector_type(4)));

__global__ __launch_bounds__(256) void elemwise_sq_affine_f32x4(
    const float* __restrict__ x, float* __restrict__ out, long long n) {
  long long i4 = (long long)blockIdx.x * blockDim.x + threadIdx.x;  // float4 index
  long long n4 = n >> 2;

  if (i4 < n4) {
    const v4f* __restrict__ xv = (const v4f*)x;
    v4f* __restrict__ ov = (v4f*)out;

    v4f v = __builtin_nontemporal_load(&xv[i4]);  // global_load_b128, TH=NT

    v4f z;
    z.x = __builtin_fmaf(0.75f, v.x, 0.25f);
    z.y = __builtin_fmaf(0.75f, v.y, 0.25f);
    z.z = __builtin_fmaf(0.75f, v.z, 0.25f);
    z.w = __builtin_fmaf(0.75f, v.w, 0.25f);
    v4f r = z * z;

    __builtin_nontemporal_store(r, &ov[i4]);  // global_store_b128, TH=NT
  }

  // Tail guard (n % 4 != 0). Dead code for n = 2^27, kept for generality.
  if (blockIdx.x == 0 && threadIdx.x == 0) {
    for (long long t = n4 << 2; t < n; ++t) {
      float z = __builtin_fmaf(0.75f, x[t], 0.25f);
      out[t] = z * z;
    }
  }
}

extern "C" void kernel_launch(void* const* d_in, const int* in_sizes, int n_in,
                              void* d_out, int out_size, void* d_ws, size_t ws_size,
                              hipStream_t stream) {
  (void)n_in; (void)d_ws; (void)ws_size; (void)out_size;
  const float* x = (const float*)d_in[0];
  float* out = (float*)d_out;

  long long n = (long long)in_sizes[0];       // 8192*16384 = 134217728
  long long n4 = n >> 2;                      // float4 count
  int block = 256;                            // 8 wave32 waves
  long long grid = (n4 + block - 1) / block;  // exact tiling (n % 1024 == 0)
  if (grid < 1) grid = 1;

  elemwise_sq_affine_f32x4<<<(dim3)(unsigned)grid, block, 0, stream>>>(x, out, n);
}